// GNNRegressor_87660282511864
// MI455X (gfx1250) — compile-verified
//
#include <hip/hip_runtime.h>
#include <hip/hip_bf16.h>
#include <math.h>

// ---------------------------------------------------------------------------
// GCN (2x GCNConv + mean-pool + MLP head) for MI455X / gfx1250.
// GEMMs use V_WMMA_F32_16X16X4_F32 (fp32 matrix cores, wave32).
// Edge scatter uses hardware global_atomic_add_f32 via __hip_atomic_fetch_add.
// ---------------------------------------------------------------------------

typedef float v2f __attribute__((ext_vector_type(2)));
typedef float v8f __attribute__((ext_vector_type(8)));

#define HID 128
#define IN_DIM 384

__device__ __forceinline__ void atomAddF(float* p, float v) {
  // Lowers to global_atomic_add_f32 (device scope, non-returning) on gfx1250.
  __hip_atomic_fetch_add(p, v, __ATOMIC_RELAXED, __HIP_MEMORY_SCOPE_AGENT);
}

// -------------------------- utility kernels --------------------------------

__global__ void k_zero(float* __restrict__ p, int n) {
  int i = blockIdx.x * blockDim.x + threadIdx.x;
  if (i < n) p[i] = 0.0f;
}

__global__ void k_deg_count(const int* __restrict__ dst, float* __restrict__ deg, int nE) {
  int i = blockIdx.x * blockDim.x + threadIdx.x;
  if (i < nE) atomAddF(&deg[dst[i]], 1.0f);
}

// dis = rsqrt(indeg + 1)   (the +1 is the self-loop; always > 0)
__global__ void k_make_dis(float* __restrict__ deg_dis, int n) {
  int i = blockIdx.x * blockDim.x + threadIdx.x;
  if (i < n) deg_dis[i] = rsqrtf(deg_dis[i] + 1.0f);
}

// -------------------------- WMMA fp32 GEMM ---------------------------------
// Y[n x 128] = X[n x K] @ W[K x 128].
// Grid.x = n/16 row tiles (n % 16 == 0). Block = 256 threads = 8 waves.
// Wave w computes the 16x16 tile at cols [16w, 16w+16). K % 4 == 0.
//
// fp32 fragment layout (ISA 7.12.2):
//  A 16x4 : lane L (m = L&15): VGPR0 = A[m][kb], VGPR1 = A[m][kb+1], kb = 2*(L>>4)
//  B 4x16 : lane L (ncol = L&15): VGPR0 = B[kb][ncol], VGPR1 = B[kb+1][ncol]
//  C 16x16: VGPR i holds C[i + 8*(L>>4)][L&15]
__global__ void k_gemm_wmma_f32(const float* __restrict__ X, const float* __restrict__ W,
                                float* __restrict__ Y, int K) {
  const int lane = threadIdx.x & 31;
  const int wave = threadIdx.x >> 5;
  const int row0 = blockIdx.x * 16;
  const int col0 = wave * 16;
  const int m    = lane & 15;          // A row / B col within the tile
  const int kb   = (lane >> 4) << 1;   // 0 or 2

  v8f c = {};
  const float* __restrict__ xrow = X + (size_t)(row0 + m) * K;
  const float* __restrict__ wcol = W + (size_t)(col0 + m);

  for (int k = 0; k < K; k += 4) {
    v2f a, b;
    a.x = xrow[k + kb];
    a.y = xrow[k + kb + 1];
    b.x = wcol[(size_t)(k + kb) * HID];
    b.y = wcol[(size_t)(k + kb + 1) * HID];
    c = __builtin_amdgcn_wmma_f32_16x16x4_f32(
        /*neg_a=*/false, a, /*neg_b=*/false, b,
        /*c_mod=*/(short)0, c, /*reuse_a=*/false, /*reuse_b=*/false);
  }

  const int rsel = (lane >> 4) * 8;
#pragma unroll
  for (int i = 0; i < 8; ++i)
    Y[(size_t)(row0 + i + rsel) * HID + col0 + m] = c[i];
}

// ----------------------- edge scatter (message passing) --------------------
// One wave per edge; 32 lanes sweep the 128 channels in 4 strided passes.
// out[dst] += h[src] * dis[src]*dis[dst]   (accumulator pre-zeroed)
__global__ void k_edge_agg(const int* __restrict__ src, const int* __restrict__ dst,
                           const float* __restrict__ dis, const float* __restrict__ h,
                           float* __restrict__ out, int nE) {
  int e = blockIdx.x * (blockDim.x >> 5) + (threadIdx.x >> 5);
  if (e >= nE) return;
  const int lane = threadIdx.x & 31;
  const int s = src[e];
  const int d = dst[e];
  const float norm = dis[s] * dis[d];
  const float* __restrict__ hs = h + (size_t)s * HID;
  float* __restrict__ od = out + (size_t)d * HID;
#pragma unroll
  for (int c = 0; c < HID; c += 32)
    atomAddF(&od[c + lane], hs[c + lane] * norm);
}

// out[i] = relu(agg[i] + dis[node]^2 * h[i] + bias[c])  -- per-element, in-place safe
__global__ void k_finish_layer(const float* __restrict__ agg, const float* __restrict__ h,
                               const float* __restrict__ dis, const float* __restrict__ bias,
                               float* __restrict__ out, int nElem) {
  int i = blockIdx.x * blockDim.x + threadIdx.x;
  if (i >= nElem) return;
  const int node = i >> 7;      // /128
  const int c    = i & (HID - 1);
  const float dii = dis[node];
  float v = agg[i] + dii * dii * h[i] + bias[c];
  out[i] = fmaxf(v, 0.0f);
}

// --------------------------- pooling + head --------------------------------

__global__ void k_pool(const float* __restrict__ h, const int* __restrict__ batch,
                       float* __restrict__ gsum, float* __restrict__ gcnt, int nElem) {
  int i = blockIdx.x * blockDim.x + threadIdx.x;
  if (i >= nElem) return;
  const int node = i >> 7;
  const int c    = i & (HID - 1);
  const int g    = batch[node];
  atomAddF(&gsum[(size_t)g * HID + c], h[i]);
  if (c == 0) atomAddF(&gcnt[g], 1.0f);
}

// One block (64 threads) per graph: mean -> relu(g@Wl1+bl1) -> sigmoid(@Wl2+bl2)
__global__ void k_head(const float* __restrict__ gsum, const float* __restrict__ gcnt,
                       const float* __restrict__ Wl1, const float* __restrict__ bl1,
                       const float* __restrict__ Wl2, const float* __restrict__ bl2,
                       float* __restrict__ out) {
  __shared__ float g[HID];
  __shared__ float hvec[64];
  const int gid = blockIdx.x;
  const int t   = threadIdx.x;

  const float inv = 1.0f / fmaxf(gcnt[gid], 1.0f);
  for (int c = t; c < HID; c += 64) g[c] = gsum[(size_t)gid * HID + c] * inv;
  __syncthreads();

  float acc = bl1[t];
  for (int k = 0; k < HID; ++k) acc = fmaf(g[k], Wl1[k * 64 + t], acc);
  hvec[t] = fmaxf(acc, 0.0f);
  __syncthreads();

  if (t < 5) {
    float a = bl2[t];
    for (int k = 0; k < 64; ++k) a = fmaf(hvec[k], Wl2[k * 5 + t], a);
    out[gid * 5 + t] = 1.0f / (1.0f + expf(-a));
  }
}

// ------------------------------ launcher -----------------------------------

extern "C" void kernel_launch(void* const* d_in, const int* in_sizes, int n_in,
                              void* d_out, int out_size, void* d_ws, size_t ws_size,
                              hipStream_t stream) {
  // setup_inputs() order:
  // 0:x [N,384] f32   1:edge_index [2,E] i32   2:batch [N] i32
  // 3:W1[384,128] 4:b1[128] 5:W2[128,128] 6:b2[128]
  // 7:Wl1[128,64] 8:bl1[64] 9:Wl2[64,5] 10:bl2[5]
  const float* x     = (const float*)d_in[0];
  const int*   eidx  = (const int*)d_in[1];
  const int*   batch = (const int*)d_in[2];
  const float* W1  = (const float*)d_in[3];
  const float* b1  = (const float*)d_in[4];
  const float* W2  = (const float*)d_in[5];
  const float* b2  = (const float*)d_in[6];
  const float* Wl1 = (const float*)d_in[7];
  const float* bl1 = (const float*)d_in[8];
  const float* Wl2 = (const float*)d_in[9];
  const float* bl2 = (const float*)d_in[10];
  float* out = (float*)d_out;

  const int nNodes  = in_sizes[0] / IN_DIM;   // 50000 (divisible by 16)
  const int nEdges  = in_sizes[1] / 2;        // 1600000
  const int nGraphs = out_size / 5;           // 512
  const int nElem   = nNodes * HID;           // 6.4M

  const int* src = eidx;
  const int* dst = eidx + nEdges;

  // workspace carve-up (256B aligned)
  char* ws = (char*)d_ws;
  size_t off = 0;
  auto carve = [&](size_t bytes) { char* p = ws + off; off = (off + bytes + 255) & ~(size_t)255; return p; };
  float* dis  = (float*)carve((size_t)nNodes * sizeof(float));           // deg -> dis (in place)
  float* bufA = (float*)carve((size_t)nElem * sizeof(float));
  float* bufB = (float*)carve((size_t)nElem * sizeof(float));
  float* gsum = (float*)carve((size_t)nGraphs * HID * sizeof(float));
  float* gcnt = (float*)carve((size_t)nGraphs * sizeof(float));
  (void)ws_size;

  const int T = 256;
  const int rowTiles  = nNodes / 16;               // 3125
  const int edgeBlocks = (nEdges + 7) / 8;         // 8 waves (edges) per block
  const int elemBlocks = (nElem + T - 1) / T;

  // --- degree / normalization -------------------------------------------------
  k_zero<<<(nNodes + T - 1) / T, T, 0, stream>>>(dis, nNodes);
  k_deg_count<<<(nEdges + T - 1) / T, T, 0, stream>>>(dst, dis, nEdges);
  k_make_dis<<<(nNodes + T - 1) / T, T, 0, stream>>>(dis, nNodes);

  // --- layer 1: h = x@W1 ; agg ; relu(agg + dis^2 h + b1) -> bufA -------------
  k_gemm_wmma_f32<<<rowTiles, T, 0, stream>>>(x, W1, bufA, IN_DIM);
  k_zero<<<elemBlocks, T, 0, stream>>>(bufB, nElem);
  k_edge_agg<<<edgeBlocks, T, 0, stream>>>(src, dst, dis, bufA, bufB, nEdges);
  k_finish_layer<<<elemBlocks, T, 0, stream>>>(bufB, bufA, dis, b1, bufA, nElem);

  // --- layer 2: h2 = bufA@W2 -> bufB ; agg -> bufA ; relu -> bufA -------------
  k_gemm_wmma_f32<<<rowTiles, T, 0, stream>>>(bufA, W2, bufB, HID);
  k_zero<<<elemBlocks, T, 0, stream>>>(bufA, nElem);
  k_edge_agg<<<edgeBlocks, T, 0, stream>>>(src, dst, dis, bufB, bufA, nEdges);
  k_finish_layer<<<elemBlocks, T, 0, stream>>>(bufA, bufB, dis, b2, bufA, nElem);

  // --- mean pool + head -------------------------------------------------------
  k_zero<<<(nGraphs * (HID + 1) + T - 1) / T, T, 0, stream>>>(gsum, nGraphs * HID);
  k_zero<<<1, T, 0, stream>>>(gcnt, nGraphs);
  k_pool<<<elemBlocks, T, 0, stream>>>(bufA, batch, gsum, gcnt, nElem);
  k_head<<<nGraphs, 64, 0, stream>>>(gsum, gcnt, Wl1, bl1, Wl2, bl2, out);
}